// GCN_30262339568316
// MI455X (gfx1250) — compile-verified
//
#include <hip/hip_runtime.h>

typedef float v2f __attribute__((ext_vector_type(2)));
typedef float v8f __attribute__((ext_vector_type(8)));

// ---------------- degree / normalization ----------------

__global__ void k_fill_deg(float* __restrict__ deg, int N) {
  int i = blockIdx.x * blockDim.x + threadIdx.x;
  if (i < N) deg[i] = 1.0f;  // self-loop
}

__global__ void k_count(const long long* __restrict__ idx, float* __restrict__ deg,
                        long long E) {
  long long e = (long long)blockIdx.x * blockDim.x + threadIdx.x;
  if (e < E) {
    long long d = idx[E + e];  // dst row of edge_index
    atomicAdd(&deg[d], 1.0f);
  }
}

__global__ void k_rsqrt(float* __restrict__ deg, int N) {
  int i = blockIdx.x * blockDim.x + threadIdx.x;
  if (i < N) {
    float v = deg[i];
    deg[i] = (v > 0.0f) ? rsqrtf(v) : 0.0f;
  }
}

// ---------------- fp32 WMMA GEMM with M-blocking + fused epilogue ----------------
// C[nrows x NOUT] = A[nrows x K] * B[K x NOUT]
// One wave computes MT row-tiles (16*MT rows) x 16 cols; B loaded once per K-step
// and reused across the MT tiles (4x less weight traffic).
// A-matrix (32-bit 16x4): lanes 0-15 hold K=k,k+1 ; lanes 16-31 hold K=k+2,k+3.
// C/D (16x16): VGPR i -> row (i + 8*hi), col = lane&15.
// Fused epilogue: stores raw C (for edge gather) and optionally
// Cs = C * dinv[row]^2 (+ bias[col]) — the self-loop-initialized accumulator.

template<int K, int NOUT, int MT>
__global__ void k_gemm_wmma(const float* __restrict__ A, const float* __restrict__ B,
                            float* __restrict__ C, float* __restrict__ Cs,
                            const float* __restrict__ dinv,
                            const float* __restrict__ bias, int nrows, int addBias) {
  const int lane  = threadIdx.x & 31;
  const int wave  = threadIdx.x >> 5;
  const int row0  = blockIdx.x * (16 * MT);
  const int col0  = wave * 16;
  const int laneM = lane & 15;
  const int hi    = lane >> 4;  // 0 or 1

  // per-tile clamped row bases (nrows is a multiple of 16)
  int rt[MT];
#pragma unroll
  for (int t = 0; t < MT; ++t) {
    int r = row0 + 16 * t;
    rt[t] = (r + 16 > nrows) ? (nrows - 16) : r;
  }

  const float* arow[MT];
#pragma unroll
  for (int t = 0; t < MT; ++t)
    arow[t] = A + (long long)(rt[t] + laneM) * K + 2 * hi;

  const float* bcol = B + (long long)(2 * hi) * NOUT + col0 + laneM;

  v8f acc[MT];
#pragma unroll
  for (int t = 0; t < MT; ++t) acc[t] = (v8f){};

#pragma unroll 4
  for (int k = 0; k < K; k += 4) {
    v2f b;
    b.x = bcol[(long long)k * NOUT];
    b.y = bcol[(long long)k * NOUT + NOUT];
#pragma unroll
    for (int t = 0; t < MT; ++t) {
      v2f a = *(const v2f*)(arow[t] + k);  // 8B-aligned (K = k+2hi, k+2hi+1)
      // 8 args: (neg_a, A, neg_b, B, c_mod, C, reuse_a, reuse_b)
      acc[t] = __builtin_amdgcn_wmma_f32_16x16x4_f32(false, a, false, b, (short)0,
                                                     acc[t], false, false);
    }
  }

  const float bval = addBias ? bias[col0 + laneM] : 0.0f;

#pragma unroll
  for (int t = 0; t < MT; ++t) {
    const long long rbase = rt[t] + 8 * hi;
    float* crow = C + rbase * NOUT + col0 + laneM;
#pragma unroll
    for (int i = 0; i < 8; ++i) {
      float v = acc[t][i];
      crow[(long long)i * NOUT] = v;
      if (Cs) {
        float dv = dinv[rbase + i];
        Cs[(rbase + i) * (long long)NOUT + col0 + laneM] = v * dv * dv + bval;
      }
    }
  }
}

// ---------------- edge scatter-add: one wave per edge ----------------

template<int F>
__global__ void k_edge_agg(const long long* __restrict__ idx,
                           const float* __restrict__ dinv,
                           const float* __restrict__ H, float* __restrict__ out,
                           long long E) {
  const int lane = threadIdx.x & 31;
  long long e = ((long long)blockIdx.x * blockDim.x + threadIdx.x) >> 5;
  if (e >= E) return;
  long long s = idx[e];
  long long d = idx[E + e];
  float nrm = dinv[s] * dinv[d];

  constexpr int FPL = F / 32;  // floats per lane (4 for F=128, 2 for F=64)
  const float* hrow = H + s * (long long)F + lane * FPL;
  float* orow = out + d * (long long)F + lane * FPL;

  float v[FPL];
#pragma unroll
  for (int j = 0; j < FPL; ++j) v[j] = hrow[j];   // contiguous -> b128/b64 load
#pragma unroll
  for (int j = 0; j < FPL; ++j) atomicAdd(&orow[j], v[j] * nrm);
}

// ---------------- bias + relu (layer 1 post-aggregation) ----------------

template<int F>
__global__ void k_bias_relu(float* __restrict__ X, const float* __restrict__ bias,
                            long long total) {
  long long i = (long long)blockIdx.x * blockDim.x + threadIdx.x;
  if (i >= total) return;
  float v = X[i] + bias[i % F];
  X[i] = v > 0.0f ? v : 0.0f;
}

// ---------------- launch ----------------

extern "C" void kernel_launch(void* const* d_in, const int* in_sizes, int n_in,
                              void* d_out, int out_size, void* d_ws, size_t ws_size,
                              hipStream_t stream) {
  const float*     x   = (const float*)d_in[0];
  const long long* idx = (const long long*)d_in[1];
  const float*     W1  = (const float*)d_in[2];
  const float*     b1  = (const float*)d_in[3];
  const float*     W2  = (const float*)d_in[4];
  const float*     b2  = (const float*)d_in[5];
  float*           out = (float*)d_out;

  const int FIN = 256, HID = 128, FOUT = 64;
  const int N = in_sizes[0] / FIN;
  const long long E = (long long)in_sizes[1] / 2;

  // workspace layout (256B aligned)
  char* ws = (char*)d_ws;
  size_t off = 0;
  auto take = [&](size_t bytes) {
    void* p = ws + off;
    off = (off + bytes + 255) & ~(size_t)255;
    return p;
  };
  float* dinv = (float*)take((size_t)N * 4);
  float* H1   = (float*)take((size_t)N * HID * 4);
  float* agg1 = (float*)take((size_t)N * HID * 4);
  float* H2   = (float*)take((size_t)N * FOUT * 4);
  (void)ws_size; (void)n_in; (void)out_size;

  const int T = 256;
  constexpr int MT = 4;  // 64 rows per workgroup-wave set

  // symmetric normalization: dinv = rsqrt(1 + indegree)
  k_fill_deg<<<(N + T - 1) / T, T, 0, stream>>>(dinv, N);
  k_count<<<(int)((E + T - 1) / T), T, 0, stream>>>(idx, dinv, E);
  k_rsqrt<<<(N + T - 1) / T, T, 0, stream>>>(dinv, N);

  // ---- layer 1: H1 = x @ W1 ; agg1 = H1*dinv^2 (fused) ; edge agg ; relu(agg1+b1)
  {
    int blocks = (N + 16 * MT - 1) / (16 * MT);
    k_gemm_wmma<256, 128, MT><<<blocks, (128 / 16) * 32, 0, stream>>>(
        x, W1, H1, agg1, dinv, nullptr, N, 0);
    k_edge_agg<128><<<(int)((E * 32 + T - 1) / T), T, 0, stream>>>(idx, dinv, H1, agg1, E);
    long long total = (long long)N * HID;
    k_bias_relu<128><<<(int)((total + T - 1) / T), T, 0, stream>>>(agg1, b1, total);
  }

  // ---- layer 2: H2 = agg1 @ W2 ; out = H2*dinv^2 + b2 (fused) ; edge agg
  {
    int blocks = (N + 16 * MT - 1) / (16 * MT);
    k_gemm_wmma<128, 64, MT><<<blocks, (64 / 16) * 32, 0, stream>>>(
        agg1, W2, H2, out, dinv, b2, N, 1);
    k_edge_agg<64><<<(int)((E * 32 + T - 1) / T), T, 0, stream>>>(idx, dinv, H2, out, E);
  }
}